// AGComplexAttention_62105227100476
// MI455X (gfx1250) — compile-verified
//
#include <hip/hip_runtime.h>
#include <hip/hip_bf16.h>

// MI455X / gfx1250 complex multi-head attention, fp32 WMMA path.
//
// Roofline: ~140 GFLOP vs ~300 MB HBM traffic -> memory-bound at 23.3 TB/s
// (~13 us mem vs ~1 us compute even at f32 WMMA rates). So we keep full fp32
// precision with V_WMMA_F32_16X16X4_F32 instead of downcasting.
//
// Attention kernel stages K/V tiles into LDS once per block via
// GLOBAL_LOAD_ASYNC_TO_LDS_B128 (ASYNCcnt), double-buffered so tile j+1
// streams in while WMMAs consume tile j.

typedef float v2f __attribute__((ext_vector_type(2)));
typedef float v8f __attribute__((ext_vector_type(8)));

#define D_MODEL 512
#define NUM_HEADS 8
#define DH 64
#define BATCH 4
#define SEQ 2048
#define M_TOTAL (BATCH * SEQ) // 8192

// D = A(16x4) * B(4x16) + C(16x16), all f32, wave32.
// A fragment (per lane): m = lane&15, k = (lane>>4)*2 + {0,1}
// B fragment (per lane): n = lane&15, k = (lane>>4)*2 + {0,1}
// C fragment (per lane): vgpr r -> m = r + (lane>>4)*8, n = lane&15
__device__ __forceinline__ v8f wmma_f32(v2f a, v2f b, v8f c) {
  return __builtin_amdgcn_wmma_f32_16x16x4_f32(
      /*neg_a=*/false, a, /*neg_b=*/false, b,
      /*c_mod=*/(short)0, c, /*reuse_a=*/false, /*reuse_b=*/false);
}

// Per-lane async copy of 16B global -> LDS (tracked by ASYNCcnt).
__device__ __forceinline__ void async_ld_b128(unsigned lds_off,
                                              const float* g) {
  asm volatile("global_load_async_to_lds_b128 %0, %1, off" ::"v"(lds_off),
               "v"((unsigned long long)(uintptr_t)g)
               : "memory");
}
__device__ __forceinline__ void wait_async0() {
  asm volatile("s_wait_asynccnt 0x0" ::: "memory");
}

// ---------------------------------------------------------------------------
// Kernel 1: Q/K/V projections.  q = x @ wq.T (complex), etc.
// blockIdx.z in {0,1,2} selects {wq,wk,wv}; outputs stored as planar re/im
// planes of shape [M_TOTAL, D_MODEL] (row-major), so head (b,h) slices are
// just base + h*DH with row stride D_MODEL.
// Each wave computes a 16(M) x 64(N) complex tile.
// ---------------------------------------------------------------------------
__global__ __launch_bounds__(128) void qkv_proj_kernel(
    const float* __restrict__ x,   // [M,512] interleaved complex (float2)
    const float* __restrict__ wq,
    const float* __restrict__ wk,
    const float* __restrict__ wv,
    float* __restrict__ ws) {
  const int lane = threadIdx.x & 31;
  const int wid = threadIdx.x >> 5;
  const int tile = blockIdx.x * 4 + wid; // 0..4095
  const int mt = tile >> 3;              // 0..511 (M tile of 16)
  const int nt = tile & 7;               // 0..7   (N tile of 64)
  const int which = blockIdx.z;

  const float* w = (which == 0) ? wq : (which == 1) ? wk : wv;
  const size_t plane = (size_t)M_TOTAL * D_MODEL;
  float* outr = ws + (size_t)which * 2 * plane;
  float* outi = outr + plane;

  const int lo = lane & 15;
  const int hi = lane >> 4;
  const int m = mt * 16 + lo; // A-fragment row
  const int khalf = hi * 2;

  v8f cr[4] = {};
  v8f ci[4] = {};

  for (int k = 0; k < D_MODEL; k += 4) {
    const int kk = k + khalf;
    // A fragment: x[m][kk], x[m][kk+1] as complex -> {re0,im0,re1,im1}
    const float4 xa = *(const float4*)(x + ((size_t)m * D_MODEL + kk) * 2);
    v2f ar = {xa.x, xa.z};
    v2f ai = {xa.y, xa.w};
    v2f ani = {-xa.y, -xa.w}; // f32 WMMA has no A-neg; negate in VALU
#pragma unroll
    for (int s = 0; s < 4; s++) {
      const int n = nt * 64 + s * 16 + lo;
      // B[k][n] = w[n][k]  (x @ w.T): rows of w, contiguous over k
      const float4 w4 = *(const float4*)(w + ((size_t)n * D_MODEL + kk) * 2);
      v2f br = {w4.x, w4.z};
      v2f bi = {w4.y, w4.w};
      // Cr += Ar*Br - Ai*Bi ;  Ci += Ar*Bi + Ai*Br
      cr[s] = wmma_f32(ar, br, cr[s]);
      cr[s] = wmma_f32(ani, bi, cr[s]);
      ci[s] = wmma_f32(ar, bi, ci[s]);
      ci[s] = wmma_f32(ai, br, ci[s]);
    }
  }

#pragma unroll
  for (int s = 0; s < 4; s++) {
    const int n = nt * 64 + s * 16 + lo;
#pragma unroll
    for (int r = 0; r < 8; r++) {
      const int mm = mt * 16 + r + hi * 8;
      outr[(size_t)mm * D_MODEL + n] = cr[s][r];
      outi[(size_t)mm * D_MODEL + n] = ci[s][r];
    }
  }
}

// ---------------------------------------------------------------------------
// Kernel 2: flash attention per (b,h).
// grid = (SEQ/64, BATCH*NUM_HEADS), block = 128 (4 waves).
// Each wave owns 16 query rows; Q fragments (pre-scaled) stay in registers.
// K/V tiles (kr/ki/vr/vi, 16x64 each) are staged to LDS once per block with
// double-buffered GLOBAL_LOAD_ASYNC_TO_LDS_B128; all 4 waves consume the
// shared tile (4x less global traffic, loads overlap WMMA).
// LDS pitch 68 floats: rows stay 16B-aligned for b128 and score-fragment
// reads hit banks (4n+d)%64 -> conflict-free.
// ---------------------------------------------------------------------------
#define KV_PITCH 68

__global__ __launch_bounds__(128) void attn_kernel(float* __restrict__ ws) {
  const size_t plane = (size_t)M_TOTAL * D_MODEL;
  const float* qr = ws;
  const float* qi = ws + plane;
  const float* kv_planes[4] = {ws + 2 * plane, ws + 3 * plane,
                               ws + 4 * plane, ws + 5 * plane};
  float* outr = ws + 6 * plane;
  float* outi = ws + 7 * plane;

  const int lane = threadIdx.x & 31;
  const int wid = threadIdx.x >> 5;
  const int lo = lane & 15;
  const int hi = lane >> 4;

  const int bh = blockIdx.y;
  const int b = bh >> 3;
  const int h = bh & 7;
  const size_t base = ((size_t)b * SEQ) * D_MODEL + (size_t)h * DH;

  const int qrow0 = blockIdx.x * 64 + wid * 16;

  __shared__ __align__(16) float tiles[2][4][16 * KV_PITCH]; // kr,ki,vr,vi
  __shared__ float stage[4][16 * 17]; // per-wave P staging, padded stride

  // Cooperative tile copy partition: 128 threads, 16 rows x (2 float4)/plane.
  const int ldr = threadIdx.x >> 3;       // row 0..15
  const int ldc = (threadIdx.x & 7) * 4;  // float col base (16B granules)

  auto issue_tile = [&](int j, int buf) {
#pragma unroll
    for (int p = 0; p < 4; ++p) {
      const float* g =
          kv_planes[p] + base + (size_t)(j * 16 + ldr) * D_MODEL + ldc;
      async_ld_b128((unsigned)(uintptr_t)&tiles[buf][p][ldr * KV_PITCH + ldc],
                    g);
      async_ld_b128(
          (unsigned)(uintptr_t)&tiles[buf][p][ldr * KV_PITCH + ldc + 32],
          g + 32);
    }
  };

  // Prologue: start streaming tile 0 while we preload Q fragments.
  issue_tile(0, 0);

  const float scale = 0.125f; // 1/sqrt(DH)

  v2f qfr[16], qfi[16];
#pragma unroll
  for (int t = 0; t < 16; t++) {
    const int d = t * 4 + hi * 2;
    const size_t a = base + (size_t)(qrow0 + lo) * D_MODEL + d;
    float2 r2 = *(const float2*)(qr + a);
    float2 i2 = *(const float2*)(qi + a);
    qfr[t] = (v2f){r2.x * scale, r2.y * scale};
    qfi[t] = (v2f){i2.x * scale, i2.y * scale};
  }

  v8f o_r[4] = {};
  v8f o_i[4] = {};
  float rmax[8], lsum[8];
#pragma unroll
  for (int r = 0; r < 8; r++) {
    rmax[r] = -3.0e38f;
    lsum[r] = 0.0f;
  }

  for (int j = 0; j < SEQ / 16; ++j) {
    const int buf = j & 1;
    // Wait for this iteration's tile (issued last iteration), make it block-
    // visible, then immediately start streaming the next tile into the other
    // buffer (safe: the barrier also proves everyone finished reading it).
    wait_async0();
    __syncthreads();
    if (j + 1 < SEQ / 16) issue_tile(j + 1, buf ^ 1);

    const float* kt_r = &tiles[buf][0][0];
    const float* kt_i = &tiles[buf][1][0];
    const float* vt_r = &tiles[buf][2][0];
    const float* vt_i = &tiles[buf][3][0];

    // ---- score tile S = scale * (qr.kr^T + qi.ki^T), 16x16 real ----
    v8f s = {};
#pragma unroll
    for (int t = 0; t < 16; t++) {
      const int d = t * 4 + hi * 2;
      float2 r2 = *(const float2*)(kt_r + lo * KV_PITCH + d); // B[d][n]=K[n][d]
      float2 i2 = *(const float2*)(kt_i + lo * KV_PITCH + d);
      s = wmma_f32(qfr[t], (v2f){r2.x, r2.y}, s);
      s = wmma_f32(qfi[t], (v2f){i2.x, i2.y}, s);
    }

    // ---- online softmax (rows live in 16-lane halves) ----
    float alpha[8];
    float psum[8];
    v8f p;
#pragma unroll
    for (int r = 0; r < 8; r++) {
      float v = s[r];
      float mx = v;
      mx = fmaxf(mx, __shfl_xor(mx, 1, 32));
      mx = fmaxf(mx, __shfl_xor(mx, 2, 32));
      mx = fmaxf(mx, __shfl_xor(mx, 4, 32));
      mx = fmaxf(mx, __shfl_xor(mx, 8, 32));
      const float nm = fmaxf(rmax[r], mx);
      alpha[r] = __expf(rmax[r] - nm);
      rmax[r] = nm;
      const float pv = __expf(v - nm);
      p[r] = pv;
      float sm = pv;
      sm += __shfl_xor(sm, 1, 32);
      sm += __shfl_xor(sm, 2, 32);
      sm += __shfl_xor(sm, 4, 32);
      sm += __shfl_xor(sm, 8, 32);
      psum[r] = sm;
    }
#pragma unroll
    for (int r = 0; r < 8; r++) lsum[r] = lsum[r] * alpha[r] + psum[r];

    // ---- rescale running O by alpha ----
#pragma unroll
    for (int dt = 0; dt < 4; dt++)
#pragma unroll
      for (int r = 0; r < 8; r++) {
        o_r[dt][r] *= alpha[r];
        o_i[dt][r] *= alpha[r];
      }

    // ---- stage P (C layout) to LDS, re-read as A fragments ----
    __syncthreads();
#pragma unroll
    for (int r = 0; r < 8; r++) stage[wid][(r + hi * 8) * 17 + lo] = p[r];
    __syncthreads();

#pragma unroll
    for (int t = 0; t < 4; t++) {
      const int kk = t * 4 + hi * 2;
      v2f pa = {stage[wid][lo * 17 + kk], stage[wid][lo * 17 + kk + 1]};
#pragma unroll
      for (int dt = 0; dt < 4; dt++) {
        const int d = dt * 16 + lo;
        v2f bvr = {vt_r[kk * KV_PITCH + d], vt_r[(kk + 1) * KV_PITCH + d]};
        v2f bvi = {vt_i[kk * KV_PITCH + d], vt_i[(kk + 1) * KV_PITCH + d]};
        o_r[dt] = wmma_f32(pa, bvr, o_r[dt]);
        o_i[dt] = wmma_f32(pa, bvi, o_i[dt]);
      }
    }
  }

  // ---- finalize: O /= l, store planar [B,S,H*DH] ----
#pragma unroll
  for (int r = 0; r < 8; r++) lsum[r] = 1.0f / lsum[r];
#pragma unroll
  for (int dt = 0; dt < 4; dt++) {
    const int d = dt * 16 + lo;
#pragma unroll
    for (int r = 0; r < 8; r++) {
      const int srow = qrow0 + r + hi * 8;
      const size_t a = base + (size_t)srow * D_MODEL + d;
      outr[a] = o_r[dt][r] * lsum[r];
      outi[a] = o_i[dt][r] * lsum[r];
    }
  }
}

// ---------------------------------------------------------------------------
// Kernel 3: final projection  out = attn_out @ wo.T  (complex), interleaved
// complex store to d_out.
// ---------------------------------------------------------------------------
__global__ __launch_bounds__(128) void out_proj_kernel(
    const float* __restrict__ ws, const float* __restrict__ wo,
    float* __restrict__ out) {
  const size_t plane = (size_t)M_TOTAL * D_MODEL;
  const float* ar_p = ws + 6 * plane;
  const float* ai_p = ws + 7 * plane;

  const int lane = threadIdx.x & 31;
  const int wid = threadIdx.x >> 5;
  const int tile = blockIdx.x * 4 + wid;
  const int mt = tile >> 3;
  const int nt = tile & 7;
  const int lo = lane & 15;
  const int hi = lane >> 4;
  const int m = mt * 16 + lo;
  const int khalf = hi * 2;

  v8f cr[4] = {};
  v8f ci[4] = {};

  for (int k = 0; k < D_MODEL; k += 4) {
    const int kk = k + khalf;
    const float2 a_r = *(const float2*)(ar_p + (size_t)m * D_MODEL + kk);
    const float2 a_i = *(const float2*)(ai_p + (size_t)m * D_MODEL + kk);
    v2f ar = {a_r.x, a_r.y};
    v2f ai = {a_i.x, a_i.y};
    v2f ani = {-a_i.x, -a_i.y};
#pragma unroll
    for (int s = 0; s < 4; s++) {
      const int n = nt * 64 + s * 16 + lo;
      const float4 w4 = *(const float4*)(wo + ((size_t)n * D_MODEL + kk) * 2);
      v2f br = {w4.x, w4.z};
      v2f bi = {w4.y, w4.w};
      cr[s] = wmma_f32(ar, br, cr[s]);
      cr[s] = wmma_f32(ani, bi, cr[s]);
      ci[s] = wmma_f32(ar, bi, ci[s]);
      ci[s] = wmma_f32(ai, br, ci[s]);
    }
  }

#pragma unroll
  for (int s = 0; s < 4; s++) {
    const int n = nt * 64 + s * 16 + lo;
#pragma unroll
    for (int r = 0; r < 8; r++) {
      const int mm = mt * 16 + r + hi * 8;
      float2 c2 = {cr[s][r], ci[s][r]};
      *(float2*)(out + ((size_t)mm * D_MODEL + n) * 2) = c2;
    }
  }
}

extern "C" void kernel_launch(void* const* d_in, const int* in_sizes, int n_in,
                              void* d_out, int out_size, void* d_ws,
                              size_t ws_size, hipStream_t stream) {
  const float* x = (const float*)d_in[0]; // complex64 interleaved
  const float* wq = (const float*)d_in[1];
  const float* wk = (const float*)d_in[2];
  const float* wv = (const float*)d_in[3];
  const float* wo = (const float*)d_in[4];
  float* ws = (float*)d_ws; // 8 planes x M_TOTAL*D_MODEL floats = 128 MiB

  // Q/K/V projections: 4096 wave-tiles per matrix, 4 waves/block, z selects W.
  dim3 g1(1024, 1, 3);
  qkv_proj_kernel<<<g1, 128, 0, stream>>>(x, wq, wk, wv, ws);

  // Flash attention: (S/64 query blocks) x (B*H heads).
  dim3 g2(SEQ / 64, BATCH * NUM_HEADS);
  attn_kernel<<<g2, 128, 0, stream>>>(ws);

  // Output projection.
  out_proj_kernel<<<1024, 128, 0, stream>>>(ws, wo, (float*)d_out);
}